// GNNLayer_29867202576483
// MI455X (gfx1250) — compile-verified
//
#include <hip/hip_runtime.h>
#include <hip/hip_bf16.h>
#include <math.h>

typedef __attribute__((ext_vector_type(2))) float v2f;
typedef __attribute__((ext_vector_type(8))) float v8f;

#define D 64

// ---------------------------------------------------------------------------
// out[n][a] = sum_d X[n][d] * W[a][d]        (X: Nrows x 64, W: 64 x 64 row-major)
// One wave (32 lanes) computes one 16x16 tile of 'out' via V_WMMA_F32_16X16X4_F32.
// Block = 128 threads = 4 waves = the 4 column tiles (64 cols) of one 16-row stripe.
//
// A (16x4 f32) layout: lane l holds row M = l%16; VGPR v holds K = (l/16)*2 + v.
// B (4x16 f32) layout (mirror): lane l holds col N = l%16; VGPR v holds same K.
//   B[k][n] = W[n][k]  -> contiguous float2 loads from W's row n.
// C/D (16x16 f32): lane l, VGPR v -> row M = (l/16)*8 + v, col N = l%16.
//
// GUARD=false: nrows is a multiple of 16, no masking at all (hot path).
// GUARD=true : branch-free tail handling — clamp the gather row to a valid
//              address and scale the A fragment by 0/1 (v_cndmask + v_mul,
//              no EXEC manipulation, so loads stay batchable).
// All 16 A / 16 B fragments are loaded up front so the compiler can issue the
// 32 global_load_b64s together and run the 16-WMMA chain behind one wait.
// ---------------------------------------------------------------------------
template <bool GUARD>
__global__ void gemm_xwT_wmma(const float* __restrict__ X,
                              const float* __restrict__ W,
                              float* __restrict__ out,
                              int nrows) {
  const int lane = threadIdx.x & 31;
  const int wave = threadIdx.x >> 5;          // 0..3 -> column tile
  const int m0   = blockIdx.x * 16;
  const int n0   = wave * 16;

  const int arow_raw = m0 + (lane & 15);      // A row this lane supplies
  const int bcol     = n0 + (lane & 15);      // B col (== W row) this lane supplies
  const int khalf    = (lane >> 4) * 2;       // K sub-offset 0 or 2

  int   arow  = arow_raw;
  float amask = 1.0f;
  if (GUARD) {
    const bool valid = arow_raw < nrows;
    arow  = valid ? arow_raw : (nrows - 1);   // clamped: always a legal address
    amask = valid ? 1.0f : 0.0f;
  }

  const float* __restrict__ aptr = X + (size_t)arow * D + khalf;
  const float* __restrict__ bptr = W + (size_t)bcol * D + khalf;

  v2f afrag[16];
  v2f bfrag[16];
#pragma unroll
  for (int j = 0; j < 16; ++j) {
    afrag[j] = *(const v2f*)(aptr + j * 4);
    bfrag[j] = *(const v2f*)(bptr + j * 4);
  }
  if (GUARD) {
#pragma unroll
    for (int j = 0; j < 16; ++j) {
      afrag[j].x *= amask;
      afrag[j].y *= amask;
    }
  }

  v8f acc = {};
#pragma unroll
  for (int j = 0; j < 16; ++j) {
    acc = __builtin_amdgcn_wmma_f32_16x16x4_f32(
        /*neg_a=*/false, afrag[j], /*neg_b=*/false, bfrag[j],
        /*c_mod=*/(short)0, acc, /*reuse_a=*/false, /*reuse_b=*/false);
  }

  const int rbase = m0 + (lane >> 4) * 8;
  const int c     = n0 + (lane & 15);
#pragma unroll
  for (int v = 0; v < 8; ++v) {
    const int r = rbase + v;
    if (!GUARD || r < nrows) out[(size_t)r * D + c] = acc[v];
  }
}

// ---------------------------------------------------------------------------
// qr_pre[b][a] = sum_d rel_emb[q_rel[b]][d] * Wqr_w[a][d] + Wqr_b[a]
// Tiny (B=8): one block, 64*B threads, one output element each.
// ---------------------------------------------------------------------------
__global__ void qr_pre_kernel(const int* __restrict__ q_rel,
                              const float* __restrict__ rel_emb,
                              const float* __restrict__ Wqr_w,
                              const float* __restrict__ Wqr_b,
                              float* __restrict__ qr_pre) {
  const int t = threadIdx.x;
  const int b = t >> 6;
  const int a = t & 63;
  const float* h = rel_emb + q_rel[b] * D;
  const float* w = Wqr_w + a * D;
  float s = Wqr_b[a];
#pragma unroll 8
  for (int d = 0; d < D; ++d) s = fmaf(h[d], w[d], s);
  qr_pre[b * D + a] = s;
}

__global__ void zero_f32(float* __restrict__ p, int n) {
  int i = blockIdx.x * blockDim.x + threadIdx.x;
  if (i < n) p[i] = 0.0f;
}

// ---------------------------------------------------------------------------
// Edge phase: one wave per edge, 2 f32 per lane (covers D=64).
//   pre   = Ws_hidden[sub] + Wr_hr[rel] + qr_pre[r_idx]
//   alpha = sigmoid(dot(relu(pre), walpha_w) + walpha_b)
//   msg_agg[obj] += alpha * (hidden[sub] + rel_emb[rel])     (f32 atomics)
// hidden (25.6MB) and Ws_hidden (25.6MB) fit in the 192MB L2, so the random
// gathers/scatters are L2-resident; HBM traffic ~ edges + node arrays.
// ---------------------------------------------------------------------------
__global__ void edge_kernel(const int* __restrict__ edges,
                            const float* __restrict__ hidden,
                            const float* __restrict__ rel_emb,
                            const float* __restrict__ ws_hidden,
                            const float* __restrict__ wr_hr,
                            const float* __restrict__ qr_pre,
                            const float* __restrict__ walpha_w,
                            const float* __restrict__ walpha_b,
                            float* __restrict__ msg_agg,
                            int n_edge) {
  const int lane = threadIdx.x & 31;
  const int wave = threadIdx.x >> 5;
  const int e = blockIdx.x * (blockDim.x >> 5) + wave;
  if (e >= n_edge) return;

  const int* ed = edges + e * 6;            // [r_idx, head, rel, tail, sub, obj]
  const int r_idx = ed[0];
  const int rel   = ed[2];
  const int sub   = ed[4];
  const int obj   = ed[5];

  const int d = lane * 2;

  v2f pre = *(const v2f*)(ws_hidden + (size_t)sub * D + d);
  v2f t1  = *(const v2f*)(wr_hr + (size_t)rel * D + d);
  v2f t2  = *(const v2f*)(qr_pre + (size_t)r_idx * D + d);
  pre.x += t1.x + t2.x;
  pre.y += t1.y + t2.y;

  v2f wa = *(const v2f*)(walpha_w + d);
  float part = fmaxf(pre.x, 0.0f) * wa.x + fmaxf(pre.y, 0.0f) * wa.y;

  // wave32 butterfly reduction
#pragma unroll
  for (int off = 16; off > 0; off >>= 1)
    part += __shfl_xor(part, off, 32);

  const float alpha = 1.0f / (1.0f + __expf(-(part + walpha_b[0])));

  v2f hs = *(const v2f*)(hidden + (size_t)sub * D + d);
  v2f hr = *(const v2f*)(rel_emb + (size_t)rel * D + d);

  float* dst = msg_agg + (size_t)obj * D + d;
  atomicAdd(dst,     alpha * (hs.x + hr.x));
  atomicAdd(dst + 1, alpha * (hs.y + hr.y));
}

// ---------------------------------------------------------------------------
// Orchestration
// ---------------------------------------------------------------------------
extern "C" void kernel_launch(void* const* d_in, const int* in_sizes, int n_in,
                              void* d_out, int out_size, void* d_ws, size_t ws_size,
                              hipStream_t stream) {
  // setup_inputs order:
  // 0 q_sub, 1 q_rel, 2 hidden, 3 edges, 4 nodes, 5 old_nodes_new_idx,
  // 6 batchsize, 7 rel_emb, 8 Ws, 9 Wr, 10 Wqr_w, 11 Wqr_b,
  // 12 walpha_w, 13 walpha_b, 14 Wh
  const int*   q_rel    = (const int*)d_in[1];
  const float* hidden   = (const float*)d_in[2];
  const int*   edges    = (const int*)d_in[3];
  const float* rel_emb  = (const float*)d_in[7];
  const float* Ws       = (const float*)d_in[8];
  const float* Wr       = (const float*)d_in[9];
  const float* Wqr_w    = (const float*)d_in[10];
  const float* Wqr_b    = (const float*)d_in[11];
  const float* walpha_w = (const float*)d_in[12];
  const float* walpha_b = (const float*)d_in[13];
  const float* Wh       = (const float*)d_in[14];

  const int n_node   = in_sizes[2] / D;
  const int n_edge   = in_sizes[3] / 6;
  const int n_reltab = in_sizes[7] / D;
  const int batch    = in_sizes[1];

  // Workspace layout (floats)
  float* ws_hidden = (float*)d_ws;                        // n_node * 64
  float* msg_agg   = ws_hidden + (size_t)n_node * D;      // n_node * 64
  float* wr_hr     = msg_agg + (size_t)n_node * D;        // n_reltab * 64
  float* qr_pre    = wr_hr + (size_t)n_reltab * D;        // batch * 64

  const bool node_mult16 = (n_node % 16) == 0;

  // 1) zero the segment-sum accumulator
  {
    const int n = n_node * D;
    zero_f32<<<(n + 255) / 256, 256, 0, stream>>>(msg_agg, n);
  }

  // 2) Ws_hidden = hidden @ Ws^T   (node-level; replaces per-edge GEMV)
  if (node_mult16)
    gemm_xwT_wmma<false><<<n_node / 16, 128, 0, stream>>>(hidden, Ws, ws_hidden, n_node);
  else
    gemm_xwT_wmma<true><<<(n_node + 15) / 16, 128, 0, stream>>>(hidden, Ws, ws_hidden, n_node);

  // 3) Wr_hr = rel_emb @ Wr^T      (401 rows, branch-free tail masking)
  gemm_xwT_wmma<true><<<(n_reltab + 15) / 16, 128, 0, stream>>>(rel_emb, Wr, wr_hr, n_reltab);

  // 4) qr_pre = rel_emb[q_rel] @ Wqr_w^T + Wqr_b   (B x 64, tiny)
  qr_pre_kernel<<<1, batch * D, 0, stream>>>(q_rel, rel_emb, Wqr_w, Wqr_b, qr_pre);

  // 5) edge phase: gather + gate + atomic scatter (1 wave per edge)
  {
    const int edges_per_block = 256 / 32;
    edge_kernel<<<(n_edge + edges_per_block - 1) / edges_per_block, 256, 0, stream>>>(
        edges, hidden, rel_emb, ws_hidden, wr_hr, qr_pre, walpha_w, walpha_b,
        msg_agg, n_edge);
  }

  // 6) hidden_new = msg_agg @ Wh^T  -> d_out
  if (node_mult16)
    gemm_xwT_wmma<false><<<n_node / 16, 128, 0, stream>>>(msg_agg, Wh, (float*)d_out, n_node);
  else
    gemm_xwT_wmma<true><<<(n_node + 15) / 16, 128, 0, stream>>>(msg_agg, Wh, (float*)d_out, n_node);
}